// GGNN1_tfidf_77764677862200
// MI455X (gfx1250) — compile-verified
//
#include <hip/hip_runtime.h>
#include <hip/hip_bf16.h>
#include <math.h>

// Problem constants (match reference)
#define D1 15
#define D2 20
#define D3 20
#define NCLS 6
#define NLAYERS 2
#define NNODES 100000      // exact multiple of 16 -> no node-tile tail
#define NEDGES 3200000
#define NGRAPHS 256
#define G3 (3 * D1)        // 45 GRU gate columns
#define PD 16              // padded feature stride (K padded 15->16)
#define PG 48              // padded gate stride (45->48)

typedef __attribute__((ext_vector_type(2))) float v2f;  // f32 WMMA A/B fragment (16x4 / 4x16)
typedef __attribute__((ext_vector_type(8))) float v8f;  // f32 WMMA C/D fragment (16x16)

// ---------------------------------------------------------------- utilities
__global__ void zero_f32(float* __restrict__ p, long n) {
  long i = (long)blockIdx.x * blockDim.x + threadIdx.x;
  if (i < n) p[i] = 0.0f;
}

// x [N][15] -> hp [N][16] zero-padded (so all WMMA A loads are unconditional b64)
__global__ void pack_x(const float* __restrict__ x, float* __restrict__ hp) {
  int i = blockIdx.x * blockDim.x + threadIdx.x;
  if (i >= NNODES * PD) return;
  const int n = i >> 4, j = i & 15;
  hp[i] = (j < D1) ? x[n * D1 + j] : 0.0f;
}

// Pre-pack all weights into zero-padded B-friendly layouts (once per launch):
//   Wt[l][n*16+k]  = weight[l][k][n]   (B^T: n-major, k contiguous)
//   wihp[j*16+k]   = w_ih[j][k],  whhp likewise  (already [N][K]); rows 45..47 zero
//   bip/bhp[48]    = biases zero-padded
__global__ void pack_params(const float* __restrict__ weight,
                            const float* __restrict__ w_ih, const float* __restrict__ w_hh,
                            const float* __restrict__ b_ih, const float* __restrict__ b_hh,
                            float* __restrict__ Wt, float* __restrict__ wihp,
                            float* __restrict__ whhp, float* __restrict__ bip,
                            float* __restrict__ bhp) {
  int i = blockIdx.x * blockDim.x + threadIdx.x;
  if (i < NLAYERS * 256) {
    const int l = i >> 8, r = i & 255, n = r >> 4, k = r & 15;
    Wt[i] = (k < D1 && n < D1) ? weight[l * D1 * D1 + k * D1 + n] : 0.0f;
  }
  const int i2 = i - NLAYERS * 256;
  if (i2 >= 0 && i2 < PG * PD) {
    const int j = i2 >> 4, k = i2 & 15;
    const bool in = (j < G3) && (k < D1);
    wihp[i2] = in ? w_ih[j * D1 + k] : 0.0f;
    whhp[i2] = in ? w_hh[j * D1 + k] : 0.0f;
  }
  const int i3 = i2 - PG * PD;
  if (i3 >= 0 && i3 < PG) {
    bip[i3] = (i3 < G3) ? b_ih[i3] : 0.0f;
    bhp[i3] = (i3 < G3) ? b_hh[i3] : 0.0f;
  }
}

// in-degree via f32 atomics (L2-resident)
__global__ void deg_kernel(const long long* __restrict__ ei, float* __restrict__ deg) {
  int e = blockIdx.x * blockDim.x + threadIdx.x;
  if (e < NEDGES) atomicAdd(&deg[(int)ei[NEDGES + e]], 1.0f);
}

__global__ void invdeg_kernel(float* __restrict__ deg) {
  int n = blockIdx.x * blockDim.x + threadIdx.x;
  if (n < NNODES) deg[n] = 1.0f / fmaxf(deg[n], 1.0f);
}

__device__ __forceinline__ v2f ld2(const float* p) {
  const float2 t = *reinterpret_cast<const float2*>(p);  // 8B-aligned b64 load
  v2f r; r[0] = t.x; r[1] = t.y; return r;
}

// ---------------------------------------------------------- m = h @ W  (padded K=16)
// One wave per 16-node tile; 4 chained f32 16x16x4 WMMAs.
// Fragment layout (ISA 7.12.2): A/B lane l, vgpr v -> element [l%16][k = v + 2*(l>>4)].
// Since k = kb + 2*half + v, the two A (and B^T) elements are CONTIGUOUS -> b64 loads.
__global__ void node_gemm_wmma(const float* __restrict__ hp, const float* __restrict__ Wt,
                               float* __restrict__ m) {
  const int wave = (blockIdx.x * blockDim.x + threadIdx.x) >> 5;
  if (wave >= NNODES / 16) return;                 // wave-uniform guard (EXEC stays full)
  const int lane = threadIdx.x & 31;
  const int half = lane >> 4, r16 = lane & 15;
  const float* arow = hp + (size_t)(wave * 16 + r16) * PD + 2 * half;  // A: M = lane%16
  const float* brow = Wt + r16 * PD + 2 * half;                        // B^T: N = lane%16
  v8f c = {};
#pragma unroll
  for (int kb = 0; kb < 16; kb += 4) {
    c = __builtin_amdgcn_wmma_f32_16x16x4_f32(false, ld2(arow + kb), false, ld2(brow + kb),
                                              (short)0, c, false, false);
  }
#pragma unroll
  for (int r = 0; r < 8; ++r) {                    // C/D: M = r + 8*half, N = lane%16
    const int M = wave * 16 + r + 8 * half;
    m[(size_t)M * PD + r16] = c[r];                // pad col (n=15) is exactly 0
  }
}

// -------------------------------------------------- edge scatter: agg[tgt] += m[src]
__global__ void scatter_kernel(const long long* __restrict__ ei,
                               const float* __restrict__ m, float* __restrict__ agg) {
  const int stride = gridDim.x * blockDim.x;
  for (int e = blockIdx.x * blockDim.x + threadIdx.x; e < NEDGES; e += stride) {
    const int ep = e + stride;                     // hint next iteration's indices
    if (ep < NEDGES) {
      __builtin_prefetch(&ei[ep], 0, 1);           // -> global_prefetch_b8
      __builtin_prefetch(&ei[NEDGES + ep], 0, 1);
    }
    const int s = (int)ei[e];
    const int t = (int)ei[NEDGES + e];
    const float4* ms = reinterpret_cast<const float4*>(m + (size_t)s * PD);  // 64B row
    const float4 r0 = ms[0], r1 = ms[1], r2 = ms[2], r3 = ms[3];
    float* at = agg + (size_t)t * PD;
    atomicAdd(&at[0],  r0.x); atomicAdd(&at[1],  r0.y);
    atomicAdd(&at[2],  r0.z); atomicAdd(&at[3],  r0.w);
    atomicAdd(&at[4],  r1.x); atomicAdd(&at[5],  r1.y);
    atomicAdd(&at[6],  r1.z); atomicAdd(&at[7],  r1.w);
    atomicAdd(&at[8],  r2.x); atomicAdd(&at[9],  r2.y);
    atomicAdd(&at[10], r2.z); atomicAdd(&at[11], r2.w);
    atomicAdd(&at[12], r3.x); atomicAdd(&at[13], r3.y);
    atomicAdd(&at[14], r3.z);                      // skip pad col 15 (stays 0)
  }
}

// ------------------- fused GRU gate GEMMs: gi = (agg*invdeg) @ w_ih^T + b_ih,
//                                           gh =  h          @ w_hh^T + b_hh
// Wave handles (node-tile t, 16-col tile jt of 48) for BOTH gi and gh: 8 WMMAs.
// Fully branch-free: padded weights/biases make pad columns evaluate to 0.
__global__ void gru_gemm_wmma(const float* __restrict__ agg, const float* __restrict__ invdeg,
                              const float* __restrict__ hp,
                              const float* __restrict__ wihp, const float* __restrict__ whhp,
                              const float* __restrict__ bip, const float* __restrict__ bhp,
                              float* __restrict__ gi, float* __restrict__ gh) {
  const int wave = (blockIdx.x * blockDim.x + threadIdx.x) >> 5;
  const int nwork = (NNODES / 16) * 3;
  if (wave >= nwork) return;                       // wave-uniform
  const int t = wave / 3, jt = wave % 3;
  const int lane = threadIdx.x & 31;
  const int half = lane >> 4, r16 = lane & 15;
  const int nrow = t * 16 + r16;
  const float sc = invdeg[nrow];                   // mean-aggregation scale fused into A
  const int j = jt * 16 + r16;                     // output gate column (N), j < 48
  const float* aArow = agg + (size_t)nrow * PD + 2 * half;
  const float* aHrow = hp + (size_t)nrow * PD + 2 * half;
  const float* bIrow = wihp + j * PD + 2 * half;
  const float* bHrow = whhp + j * PD + 2 * half;
  v8f ci = {}, ch = {};
#pragma unroll
  for (int kb = 0; kb < 16; kb += 4) {
    v2f aA = ld2(aArow + kb);
    aA[0] *= sc; aA[1] *= sc;
    const v2f aH = ld2(aHrow + kb);
    ci = __builtin_amdgcn_wmma_f32_16x16x4_f32(false, aA, false, ld2(bIrow + kb),
                                               (short)0, ci, false, false);
    ch = __builtin_amdgcn_wmma_f32_16x16x4_f32(false, aH, false, ld2(bHrow + kb),
                                               (short)0, ch, false, false);
  }
  const float bi = bip[j], bh = bhp[j];
#pragma unroll
  for (int r = 0; r < 8; ++r) {
    const int M = t * 16 + r + 8 * half;
    gi[(size_t)M * PG + j] = ci[r] + bi;           // pad cols 45..47 hold zeros
    gh[(size_t)M * PG + j] = ch[r] + bh;
  }
}

// ------------------------------------------------------------ GRU elementwise update
__global__ void gru_elem_kernel(const float* __restrict__ gi, const float* __restrict__ gh,
                                const float* __restrict__ hp, float* __restrict__ hn) {
  int n = blockIdx.x * blockDim.x + threadIdx.x;
  if (n >= NNODES) return;
  const float* Gi = gi + (size_t)n * PG;
  const float* Gh = gh + (size_t)n * PG;
  const float* H = hp + (size_t)n * PD;
  float* O = hn + (size_t)n * PD;
#pragma unroll
  for (int c = 0; c < D1; ++c) {
    const float r = 1.0f / (1.0f + expf(-(Gi[c] + Gh[c])));
    const float z = 1.0f / (1.0f + expf(-(Gi[D1 + c] + Gh[D1 + c])));
    const float nn = tanhf(Gi[2 * D1 + c] + r * Gh[2 * D1 + c]);
    O[c] = (1.0f - z) * nn + z * H[c];
  }
  O[D1] = 0.0f;                                    // keep pad column zero
}

// ------------------------------------------- relu + mean pooling over batch_vector
__global__ void pool_kernel(const float* __restrict__ hp, const long long* __restrict__ batch,
                            float* __restrict__ pooled, float* __restrict__ cnt) {
  int n = blockIdx.x * blockDim.x + threadIdx.x;
  if (n >= NNODES) return;
  const int g = (int)batch[n];
  atomicAdd(&cnt[g], 1.0f);
#pragma unroll
  for (int j = 0; j < D1; ++j)
    atomicAdd(&pooled[g * D1 + j], fmaxf(hp[(size_t)n * PD + j], 0.0f));
}

// ---------------------------------------------- MLP head + log_softmax (256 graphs)
__global__ void head_kernel(const float* __restrict__ pooled, const float* __restrict__ cnt,
                            const float* __restrict__ tfidf,
                            const float* __restrict__ fc1_w, const float* __restrict__ fc1_b,
                            const float* __restrict__ fc2_w, const float* __restrict__ fc2_b,
                            float* __restrict__ out) {
  int g = blockIdx.x * blockDim.x + threadIdx.x;
  if (g >= NGRAPHS) return;
  float feat[D1 + D2];
  const float ic = 1.0f / fmaxf(cnt[g], 1.0f);
  for (int j = 0; j < D1; ++j) feat[j] = pooled[g * D1 + j] * ic;
  for (int j = 0; j < D2; ++j) feat[D1 + j] = tfidf[g * D2 + j];
  float o1[D3];
  for (int i = 0; i < D3; ++i) {
    float s = fc1_b[i];
    for (int k = 0; k < D1 + D2; ++k) s += fc1_w[i * (D1 + D2) + k] * feat[k];
    o1[i] = fmaxf(s, 0.0f);
  }
  float o2[NCLS], mx = -3.0e38f;
  for (int c = 0; c < NCLS; ++c) {
    float s = fc2_b[c];
    for (int i = 0; i < D3; ++i) s += fc2_w[c * D3 + i] * o1[i];
    o2[c] = s; mx = fmaxf(mx, s);
  }
  float se = 0.0f;
  for (int c = 0; c < NCLS; ++c) se += expf(o2[c] - mx);
  const float ls = logf(se);
  for (int c = 0; c < NCLS; ++c) out[g * NCLS + c] = o2[c] - mx - ls;
}

// --------------------------------------------------------------------------- driver
extern "C" void kernel_launch(void* const* d_in, const int* in_sizes, int n_in,
                              void* d_out, int out_size, void* d_ws, size_t ws_size,
                              hipStream_t stream) {
  const float*     x      = (const float*)d_in[0];
  const long long* ei     = (const long long*)d_in[1];
  const long long* batch  = (const long long*)d_in[2];
  const float*     tfidf  = (const float*)d_in[3];
  const float*     weight = (const float*)d_in[4];
  const float*     w_ih   = (const float*)d_in[5];
  const float*     w_hh   = (const float*)d_in[6];
  const float*     b_ih   = (const float*)d_in[7];
  const float*     b_hh   = (const float*)d_in[8];
  const float*     fc1_w  = (const float*)d_in[9];
  const float*     fc1_b  = (const float*)d_in[10];
  const float*     fc2_w  = (const float*)d_in[11];
  const float*     fc2_b  = (const float*)d_in[12];
  float* out = (float*)d_out;

  // Workspace layout (~71 MB of f32); every region stays 64B aligned
  float* ws     = (float*)d_ws;
  float* invdeg = ws; ws += NNODES;
  float* hp0    = ws; ws += (size_t)NNODES * PD;   // packed x
  float* m      = ws; ws += (size_t)NNODES * PD;
  float* agg    = ws; ws += (size_t)NNODES * PD;
  float* gi     = ws; ws += (size_t)NNODES * PG;
  float* gh     = ws; ws += (size_t)NNODES * PG;
  float* h1     = ws; ws += (size_t)NNODES * PD;
  float* h2     = ws; ws += (size_t)NNODES * PD;
  float* pooled = ws; ws += NGRAPHS * D1;          // pooled and cnt contiguous
  float* cnt    = ws; ws += NGRAPHS;
  float* Wt     = ws; ws += NLAYERS * 256;         // packed per-layer B^T
  float* wihp   = ws; ws += PG * PD;
  float* whhp   = ws; ws += PG * PD;
  float* bip    = ws; ws += PG;
  float* bhp    = ws; ws += PG;
  (void)ws_size; (void)in_sizes; (void)n_in; (void)out_size;

  const int B = 256;
  // packing + in-degree -> clamped reciprocal
  pack_x<<<(NNODES * PD + B - 1) / B, B, 0, stream>>>(x, hp0);
  pack_params<<<(NLAYERS * 256 + PG * PD + PG + B - 1) / B, B, 0, stream>>>(
      weight, w_ih, w_hh, b_ih, b_hh, Wt, wihp, whhp, bip, bhp);
  zero_f32<<<(NNODES + B - 1) / B, B, 0, stream>>>(invdeg, NNODES);
  deg_kernel<<<(NEDGES + B - 1) / B, B, 0, stream>>>(ei, invdeg);
  invdeg_kernel<<<(NNODES + B - 1) / B, B, 0, stream>>>(invdeg);

  const float* hcur = hp0;                         // never mutate inputs
  float* hbufs[2] = { h1, h2 };
  for (int layer = 0; layer < NLAYERS; ++layer) {
    float* hnext = hbufs[layer];
    node_gemm_wmma<<<(NNODES / 16 + 7) / 8, B, 0, stream>>>(hcur, Wt + layer * 256, m);
    zero_f32<<<(int)(((long)NNODES * PD + B - 1) / B), B, 0, stream>>>(agg, (long)NNODES * PD);
    scatter_kernel<<<4096, B, 0, stream>>>(ei, m, agg);
    const int nwork = (NNODES / 16) * 3;
    gru_gemm_wmma<<<(nwork + 7) / 8, B, 0, stream>>>(agg, invdeg, hcur, wihp, whhp, bip, bhp, gi, gh);
    gru_elem_kernel<<<(NNODES + B - 1) / B, B, 0, stream>>>(gi, gh, hcur, hnext);
    hcur = hnext;
  }

  zero_f32<<<(NGRAPHS * (D1 + 1) + B - 1) / B, B, 0, stream>>>(pooled, NGRAPHS * (D1 + 1));
  pool_kernel<<<(NNODES + B - 1) / B, B, 0, stream>>>(hcur, batch, pooled, cnt);
  head_kernel<<<1, B, 0, stream>>>(pooled, cnt, tfidf, fc1_w, fc1_b, fc2_w, fc2_b, out);
}